// GCommNetMLP_59949153517804
// MI455X (gfx1250) — compile-verified
//
#include <hip/hip_runtime.h>
#include <math.h>
#include <stdint.h>

#define NAGENT 256
#define HID    1024
#define NINP   512
#define NACT   8
#define NE     65280   // 256 * 255 directed edges

typedef float v2f __attribute__((ext_vector_type(2)));
typedef float v8f __attribute__((ext_vector_type(8)));

static __device__ __forceinline__ v8f wmma_f32(v2f a, v2f b, v8f c) {
  // V_WMMA_F32_16X16X4_F32: D = A(16x4) * B(4x16) + C(16x16), fp32 throughout.
  return __builtin_amdgcn_wmma_f32_16x16x4_f32(false, a, false, b, (short)0, c,
                                               false, false);
}

// ---------------------------------------------------------------------------
// CDNA5 async global->LDS copy (GLOBAL_LOAD_ASYNC_TO_LDS_B128, ASYNCcnt).
// LDS operand is the wave-relative LDS byte address = low 32 bits of the
// generic pointer (flat->LDS translation truncates to addr[31:0]).
// ---------------------------------------------------------------------------
static __device__ __forceinline__ void async_g2l_b128(const float* g, const float* l) {
  const uint32_t lds = (uint32_t)(uintptr_t)l;
  const uint64_t ga  = (uint64_t)(uintptr_t)g;
  asm volatile("global_load_async_to_lds_b128 %0, %1, off"
               :: "v"(lds), "v"(ga) : "memory");
}
static __device__ __forceinline__ void wait_async0() {
#if __has_builtin(__builtin_amdgcn_s_wait_asynccnt)
  __builtin_amdgcn_s_wait_asynccnt(0);
#else
  asm volatile("s_wait_asynccnt 0" ::: "memory");
#endif
}

// ---------------------------------------------------------------------------
// Generic fp32 WMMA GEMM: C[M,N] = epilogue(A[M,K] @ B[K,N])
//   mode 0: C = acc
//   mode 1: C = tanh(acc + bias[n])          (+ optional duplicate write C2)
//   mode 2: C = acc + bias[n] + D[m,n]       (GCN: Adj@Wx + Wx + b)
// Block tile 32(M) x 64(N), BK=16. 128 threads = 4 waves (2M x 2N); each wave
// owns TWO independent 16x16 fp32 accumulators (16x32 output) so the XDL pipe
// sees two interleaved accumulation chains and the A fragment is reused.
// Double-buffered LDS: the next chunk's async DMA overlaps the current chunk's
// WMMA; one s_wait_asynccnt + one barrier per chunk. All K are even multiples
// of BK (256/512/1024 -> 16/32/64 chunks), so the pipeline is a clean 2x unroll.
// ---------------------------------------------------------------------------
#define BM 32
#define BN 64
#define BK 16
#define ASTR 20   // 80B rows  -> 16B aligned, conflict-free frag reads
#define BSTR 72   // 288B rows -> 16B aligned, conflict-free frag reads
#define TPB 128

__global__ __launch_bounds__(TPB)
void gemm_wmma_f32(const float* __restrict__ A, const float* __restrict__ B,
                   float* __restrict__ C, const float* __restrict__ bias,
                   const float* __restrict__ D, float* __restrict__ C2,
                   int M, int N, int K, int mode)
{
  __shared__ float As[2][BM * ASTR];
  __shared__ float Bs[2][BK * BSTR];

  const int tid  = threadIdx.x;
  const int lane = tid & 31;
  const int wave = tid >> 5;    // 0..3
  const int wm   = wave & 1;    // M tile (0..1)
  const int wn   = wave >> 1;   // N strip of 32 (0..1)
  const int m0   = blockIdx.y * BM;
  const int n0   = blockIdx.x * BN;

  const int lrow = lane & 15;          // matrix row (A) / col (B) for this lane
  const int lk   = (lane >> 4) << 1;   // K sub-offset: 0 or 2

  // global->LDS staging indices (128 threads)
  const int ar = tid >> 2;             // 0..31  (A row)
  const int ac = (tid & 3) << 2;       // 0,4,8,12 (A col, float4)
  const int br = tid >> 3;             // 0..15  (B row)
  const int bc = (tid & 7) << 3;       // 0..56  (B col, 2x float4)

  v8f acc0 = {};
  v8f acc1 = {};

  auto stage = [&](int buf, int k0) {
    const float* gA  = &A[(size_t)(m0 + ar) * K + k0 + ac];
    const float* gB0 = &B[(size_t)(k0 + br) * N + n0 + bc];
    async_g2l_b128(gA,      &As[buf][ar * ASTR + ac]);
    async_g2l_b128(gB0,     &Bs[buf][br * BSTR + bc]);
    async_g2l_b128(gB0 + 4, &Bs[buf][br * BSTR + bc + 4]);
  };

  auto compute = [&](const float* as, const float* bs) {
#pragma unroll
    for (int ks = 0; ks < BK; ks += 4) {
      v2f a, b0, b1;
      const float* ap = &as[(wm * 16 + lrow) * ASTR + ks + lk];
      a.x = ap[0];
      a.y = ap[1];
      const float* bp = &bs[(ks + lk) * BSTR + wn * 32 + lrow];
      b0.x = bp[0];
      b0.y = bp[BSTR];
      b1.x = bp[16];
      b1.y = bp[BSTR + 16];
      acc0 = wmma_f32(a, b0, acc0);
      acc1 = wmma_f32(a, b1, acc1);
    }
  };

  // software pipeline: prefetch chunk i+1 (async DMA) while WMMA-ing chunk i
  stage(0, 0);
  wait_async0();
  __syncthreads();
  for (int k0 = 0; k0 < K; k0 += 2 * BK) {
    stage(1, k0 + BK);
    compute(As[0], Bs[0]);
    wait_async0();
    __syncthreads();
    if (k0 + 2 * BK < K) stage(0, k0 + 2 * BK);
    compute(As[1], Bs[1]);
    wait_async0();
    __syncthreads();
  }

  // C/D layout: VGPR r -> row (r | +8 for upper half-wave), col = lane&15
  const int mr0 = m0 + wm * 16 + ((lane >> 4) << 3);
  const int nc  = n0 + wn * 32 + lrow;
#pragma unroll
  for (int r = 0; r < 8; ++r) {
    const int mr = mr0 + r;
    float v0 = acc0[r];
    float v1 = acc1[r];
    if (mode == 1) {
      v0 = tanhf(v0 + bias[nc]);
      v1 = tanhf(v1 + bias[nc + 16]);
    } else if (mode == 2) {
      v0 += bias[nc]      + D[(size_t)mr * N + nc];
      v1 += bias[nc + 16] + D[(size_t)mr * N + nc + 16];
    }
    C[(size_t)mr * N + nc]      = v0;
    C[(size_t)mr * N + nc + 16] = v1;
    if (C2) {
      C2[(size_t)mr * N + nc]      = v0;
      C2[(size_t)mr * N + nc + 16] = v1;
    }
  }
}

// ---------------------------------------------------------------------------
// Per-edge gate: w[e] = ((relu(ps[src]+pd[dst]+b1) . (W2[:,1]-W2[:,0]))
//                        + (b2[1]-b2[0]) + (g1-g0)) > 0  ? 1 : 0
// (straight-through gumbel argmax is binary in the forward pass)
// One wave per edge; 8 edges per 256-thread block.
// ---------------------------------------------------------------------------
__global__ __launch_bounds__(256)
void edge_gate_kernel(const float* __restrict__ ps, const float* __restrict__ pd,
                      const float* __restrict__ b1, const float* __restrict__ W2,
                      const float* __restrict__ b2, const float* __restrict__ gum,
                      float* __restrict__ w_edge)
{
  const int wid  = (blockIdx.x << 3) + (threadIdx.x >> 5);
  const int lane = threadIdx.x & 31;
  if (wid >= NE) return;
  const int src = wid / 255;
  const int j   = wid % 255;
  const int dst = j + (j >= src);      // skip the diagonal
  const float* prow = ps + (size_t)src * (HID / 2);
  const float* qrow = pd + (size_t)dst * (HID / 2);
  float partial = 0.f;
  for (int k = lane; k < HID / 2; k += 32) {
    float h = prow[k] + qrow[k] + b1[k];
    h = h > 0.f ? h : 0.f;
    partial += h * (W2[2 * k + 1] - W2[2 * k]);
  }
#pragma unroll
  for (int off = 16; off; off >>= 1) partial += __shfl_xor(partial, off, 32);
  if (lane == 0) {
    const float ldiff = partial + (b2[1] - b2[0]);
    const float g0 = gum[2 * (size_t)wid], g1 = gum[2 * (size_t)wid + 1];
    w_edge[wid] = (ldiff + (g1 - g0)) > 0.f ? 1.f : 0.f;
  }
}

// Adj[t, s] = w_edge(s -> t), 0 on diagonal. Edge (s, t) index = s*255 + (t<s ? t : t-1)
__global__ __launch_bounds__(256)
void build_adj_kernel(const float* __restrict__ w_edge, float* __restrict__ Adj)
{
  const int idx = blockIdx.x * 256 + threadIdx.x;   // 65536 entries
  const int t = idx >> 8, s = idx & 255;
  float v = 0.f;
  if (s != t) v = w_edge[s * 255 + (t < s ? t : t - 1)];
  Adj[idx] = v;
}

// hidden = tanh(x_enc + t1 + bf[n] + t2 + bc[n])
__global__ __launch_bounds__(256)
void combine_kernel(const float* __restrict__ x_enc, const float* __restrict__ t1,
                    const float* __restrict__ bf, const float* __restrict__ t2,
                    const float* __restrict__ bc, float* __restrict__ hidden)
{
  const int idx = blockIdx.x * 256 + threadIdx.x;   // 256*1024 elements
  const int n = idx & (HID - 1);
  hidden[idx] = tanhf(x_enc[idx] + t1[idx] + bf[n] + t2[idx] + bc[n]);
}

// Heads: a_mean = h@Wa + ba, value = h@Wv + bv, log_std broadcast, std = exp.
// One wave per agent row. Output layout: [a_mean 2048 | a_log_std 2048 | a_std 2048 | value 256]
__global__ __launch_bounds__(256)
void heads_kernel(const float* __restrict__ hidden, const float* __restrict__ Wv,
                  const float* __restrict__ bv, const float* __restrict__ Wa,
                  const float* __restrict__ ba, const float* __restrict__ log_std,
                  float* __restrict__ out)
{
  const int row  = (blockIdx.x << 3) + (threadIdx.x >> 5);
  const int lane = threadIdx.x & 31;
  if (row >= NAGENT) return;
  const float* h = hidden + (size_t)row * HID;
  float pv = 0.f;
  float pa[NACT];
#pragma unroll
  for (int jj = 0; jj < NACT; ++jj) pa[jj] = 0.f;
  for (int k = lane; k < HID; k += 32) {
    const float hv = h[k];
    pv += hv * Wv[k];
#pragma unroll
    for (int jj = 0; jj < NACT; ++jj) pa[jj] += hv * Wa[k * NACT + jj];
  }
#pragma unroll
  for (int off = 16; off; off >>= 1) {
    pv += __shfl_xor(pv, off, 32);
#pragma unroll
    for (int jj = 0; jj < NACT; ++jj) pa[jj] += __shfl_xor(pa[jj], off, 32);
  }
  if (lane == 0) {
    out[3 * NAGENT * NACT + row] = pv + bv[0];          // value
#pragma unroll
    for (int jj = 0; jj < NACT; ++jj)
      out[row * NACT + jj] = pa[jj] + ba[jj];           // a_mean
  }
  if (lane < NACT) {
    const float ls = log_std[lane];
    out[NAGENT * NACT     + row * NACT + lane] = ls;       // a_log_std
    out[2 * NAGENT * NACT + row * NACT + lane] = expf(ls); // a_std
  }
}

// ---------------------------------------------------------------------------
extern "C" void kernel_launch(void* const* d_in, const int* in_sizes, int n_in,
                              void* d_out, int out_size, void* d_ws, size_t ws_size,
                              hipStream_t stream)
{
  (void)in_sizes; (void)n_in; (void)out_size; (void)ws_size;

  const float* x      = (const float*)d_in[0];
  const float* gumbel = (const float*)d_in[1];   // [P,2,E,2]
  const float* W_enc  = (const float*)d_in[2];
  const float* b_enc  = (const float*)d_in[3];
  const float* ha1_W1 = (const float*)d_in[4];   // [2H, H/2]
  const float* ha1_b1 = (const float*)d_in[5];
  const float* ha1_W2 = (const float*)d_in[6];   // [H/2, 2]
  const float* ha1_b2 = (const float*)d_in[7];
  const float* ha2_W1 = (const float*)d_in[8];
  const float* ha2_b1 = (const float*)d_in[9];
  const float* ha2_W2 = (const float*)d_in[10];
  const float* ha2_b2 = (const float*)d_in[11];
  const float* g1_W   = (const float*)d_in[12];
  const float* g1_b   = (const float*)d_in[13];
  const float* g2_W   = (const float*)d_in[14];
  const float* g2_b   = (const float*)d_in[15];
  const float* Wf     = (const float*)d_in[16];  // [P,H,H]
  const float* bf     = (const float*)d_in[17];  // [P,H]
  const float* Wc     = (const float*)d_in[18];
  const float* bc     = (const float*)d_in[19];
  const float* Wv     = (const float*)d_in[20];
  const float* bv     = (const float*)d_in[21];
  const float* Wa     = (const float*)d_in[22];
  const float* ba     = (const float*)d_in[23];
  const float* log_std= (const float*)d_in[24];

  float* ws      = (float*)d_ws;
  float* x_enc   = ws;                       // 256*1024
  float* hidden  = x_enc  + NAGENT * HID;    // 256*1024
  float* comm    = hidden + NAGENT * HID;    // 256*1024
  float* t1      = comm   + NAGENT * HID;    // 256*1024
  float* t2      = t1     + NAGENT * HID;    // 256*1024
  float* ps      = t2     + NAGENT * HID;    // 256*512
  float* pd      = ps     + NAGENT * (HID/2);// 256*512
  float* w_edge  = pd     + NAGENT * (HID/2);// 65536 (NE used)
  float* Adj     = w_edge + 65536;           // 256*256

  const dim3 blk256(256);
  const dim3 blkG(TPB);
  auto gemm = [&](const float* A, const float* B, float* C, const float* bias,
                  const float* D, float* C2, int M, int N, int K, int mode) {
    dim3 grid(N / BN, M / BM);
    gemm_wmma_f32<<<grid, blkG, 0, stream>>>(A, B, C, bias, D, C2, M, N, K, mode);
  };

  // encoder: x_enc = tanh(x @ W_enc + b_enc); hidden = x_enc (dual write)
  gemm(x, W_enc, x_enc, b_enc, nullptr, hidden, NAGENT, HID, NINP, 1);

  for (int i = 0; i < 2; ++i) {
    // --- edge gate 1 on `hidden` ---
    gemm(hidden, ha1_W1,                           ps, nullptr, nullptr, nullptr,
         NAGENT, HID / 2, HID, 0);
    gemm(hidden, ha1_W1 + (size_t)HID * (HID / 2), pd, nullptr, nullptr, nullptr,
         NAGENT, HID / 2, HID, 0);
    edge_gate_kernel<<<NE / 8, blk256, 0, stream>>>(
        ps, pd, ha1_b1, ha1_W2, ha1_b2,
        gumbel + (size_t)(i * 2 + 0) * NE * 2, w_edge);
    build_adj_kernel<<<256, blk256, 0, stream>>>(w_edge, Adj);
    // --- GCN 1: comm = Adj @ (hidden@g1_W) + (hidden@g1_W) + g1_b ---
    gemm(hidden, g1_W, t1, nullptr, nullptr, nullptr, NAGENT, HID, HID, 0);
    gemm(Adj, t1, comm, g1_b, t1, nullptr, NAGENT, HID, NAGENT, 2);

    // --- edge gate 2 on `comm` ---
    gemm(comm, ha2_W1,                             ps, nullptr, nullptr, nullptr,
         NAGENT, HID / 2, HID, 0);
    gemm(comm, ha2_W1 + (size_t)HID * (HID / 2),   pd, nullptr, nullptr, nullptr,
         NAGENT, HID / 2, HID, 0);
    edge_gate_kernel<<<NE / 8, blk256, 0, stream>>>(
        ps, pd, ha2_b1, ha2_W2, ha2_b2,
        gumbel + (size_t)(i * 2 + 1) * NE * 2, w_edge);
    build_adj_kernel<<<256, blk256, 0, stream>>>(w_edge, Adj);
    // --- GCN 2: comm = Adj @ (comm@g2_W) + (comm@g2_W) + g2_b ---
    gemm(comm, g2_W, t1, nullptr, nullptr, nullptr, NAGENT, HID, HID, 0);
    gemm(Adj, t1, comm, g2_b, t1, nullptr, NAGENT, HID, NAGENT, 2);

    // --- recurrent update: hidden = tanh(x_enc + hidden@Wf[i] + bf[i] + comm@Wc[i] + bc[i]) ---
    gemm(hidden, Wf + (size_t)i * HID * HID, t1, nullptr, nullptr, nullptr,
         NAGENT, HID, HID, 0);
    gemm(comm,   Wc + (size_t)i * HID * HID, t2, nullptr, nullptr, nullptr,
         NAGENT, HID, HID, 0);
    combine_kernel<<<(NAGENT * HID) / 256, blk256, 0, stream>>>(
        x_enc, t1, bf + (size_t)i * HID, t2, bc + (size_t)i * HID, hidden);
  }

  heads_kernel<<<NAGENT / 8, blk256, 0, stream>>>(hidden, Wv, bv, Wa, ba, log_std,
                                                  (float*)d_out);
}